// PinnGenerator_80066780332267
// MI455X (gfx1250) — compile-verified
//
#include <hip/hip_runtime.h>
#include <cstdint>
#include <cstddef>

// ---------------------------------------------------------------------------
// Jacobi relaxation of variable-coefficient 5-point stencil (PINN generator).
// G=1024 grid, B=8 batches, 65 Jacobi sweeps, fp32.
// Bandwidth-bound (AI ~1.2 flop/byte); working set 134MB < 192MB L2.
// gfx1250 path: TDM tensor_load_to_lds 2D tile DMA -> LDS halo tiles,
// s_wait_tensorcnt, wave32 64x16 tiling.
// ---------------------------------------------------------------------------

#define GDIM   1024
#define GINT   1022                 // interior size
#define TX     64                   // tile interior cols
#define TY     16                   // tile interior rows
#define HTX    (TX + 2)             // 66 (halo)
#define HTY    (TY + 2)             // 18 (halo)
#define NTHR   256                  // 8 waves (wave32)
#define MU_F   0.1f

typedef unsigned int v4u __attribute__((ext_vector_type(4)));
typedef int          v8i __attribute__((ext_vector_type(8)));
typedef int          v4i __attribute__((ext_vector_type(4)));

#if defined(__HIP_DEVICE_COMPILE__) && !__has_builtin(__builtin_amdgcn_tensor_load_to_lds)
#error "CDNA5 probe: __builtin_amdgcn_tensor_load_to_lds not available on this toolchain"
#endif

__device__ __forceinline__ int imin(int a, int b) { return a < b ? a : b; }

#if defined(__HIP_DEVICE_COMPILE__)
// Issue one TDM 2D tile load: HTY rows x HTX fp32 elements, global row stride
// GDIM elements, packed contiguously into LDS at lds_byte_off.
__device__ __forceinline__ void tdm_load_tile_f32(unsigned lds_byte_off,
                                                  const float* tile_start)
{
    unsigned long long ga = (unsigned long long)(uintptr_t)tile_start;

    // D# group 0 (128b): count=1 | lds_addr | global_addr[56:0] | type=2
    v4u g0;
    g0.x = 1u;                                            // count=1, user mode
    g0.y = lds_byte_off;                                  // LDS dest (bytes)
    g0.z = (unsigned)(ga & 0xFFFFFFFFull);                // global_addr[31:0]
    g0.w = (unsigned)((ga >> 32) & 0x01FFFFFFull)         // global_addr[56:32]
         | (2u << 30);                                    // type = 2 ("image")

    // D# group 1 (256b): data_size=4B, tensor dims huge (no OOB clamping --
    // tiles are origin-clamped so every read is in-bounds), tile 66x18,
    // tensor_dim0_stride = 1024 elements.
    v8i g1;
    g1[0] = (int)(2u << 16);                 // data_size=2 (4 bytes)
    g1[1] = (int)0xFFFF0000u;                // tensor_dim0[15:0]=0xFFFF @ [31:16]
    g1[2] = (int)0xFFFF7FFFu;                // tensor_dim0[31:16]=0x7FFF, tensor_dim1[15:0]=0xFFFF
    g1[3] = (int)(((unsigned)HTX << 16) | 0x7FFFu); // tensor_dim1 hi, tile_dim0=66
    g1[4] = (int)HTY;                        // tile_dim1=18, tile_dim2=0
    g1[5] = (int)GDIM;                       // tensor_dim0_stride[31:0] = 1024
    g1[6] = 0;                               // stride hi / dim1_stride lo
    g1[7] = 0;

    v4i gz = {0, 0, 0, 0};                   // groups 2/3 unused (2D tensor)
#if __clang_major__ >= 23
    v8i gz8 = {0, 0, 0, 0, 0, 0, 0, 0};
    __builtin_amdgcn_tensor_load_to_lds(g0, g1, gz, gz, gz8, 0);
#else
    __builtin_amdgcn_tensor_load_to_lds(g0, g1, gz, gz, 0);
#endif
}
#endif // __HIP_DEVICE_COMPILE__

// ---------------------------------------------------------------------------
// seed: zero both padded ping-pong u buffers (ring stays 0 == GD forever),
// copy `pre` into the interior of u0. Re-run every launch -> deterministic.
// ---------------------------------------------------------------------------
__global__ __launch_bounds__(NTHR)
void seed_kernel(const float* __restrict__ pre,
                 float* __restrict__ u0, float* __restrict__ u1)
{
    int idx = (int)blockIdx.x * NTHR + (int)threadIdx.x;   // 0 .. 8*1024*1024-1
    int b  = idx >> 20;
    int rc = idx & (GDIM * GDIM - 1);
    int r  = rc >> 10;
    int c  = rc & (GDIM - 1);
    float v = 0.0f;
    if (r >= 1 && r <= GINT && c >= 1 && c <= GINT)
        v = pre[(size_t)b * GINT * GINT + (size_t)(r - 1) * GINT + (c - 1)];
    u0[idx] = v;
    u1[idx] = 0.0f;
}

// ---------------------------------------------------------------------------
// One Jacobi sweep. Block = 64x16 interior tile, 256 threads, 4 pts/thread.
// ---------------------------------------------------------------------------
__global__ __launch_bounds__(NTHR)
void jac_kernel(const float* __restrict__ w, const float* __restrict__ f,
                const float* __restrict__ uin, float* __restrict__ out,
                int last)
{
    __shared__ float su_[HTY * HTX];   // u halo tile
    __shared__ float sk_[HTY * HTX];   // w tile -> converted to wk = MU + w^2

    const int b  = (int)blockIdx.z;
    const int i0 = imin((int)blockIdx.y * TY, GINT - TY);  // clamped tile origin
    const int j0 = imin((int)blockIdx.x * TX, GINT - TX);
    const size_t plane = (size_t)GDIM * GDIM;

    const float* ut = uin + (size_t)b * plane + (size_t)i0 * GDIM + j0;
    const float* wt = w   + (size_t)b * plane + (size_t)i0 * GDIM + j0;

#if defined(__HIP_DEVICE_COMPILE__)
    if (threadIdx.x < 32) {            // wave 0 issues both tile DMAs
        tdm_load_tile_f32((unsigned)(uintptr_t)&su_[0], ut);
        tdm_load_tile_f32((unsigned)(uintptr_t)&sk_[0], wt);
        __builtin_amdgcn_s_wait_tensorcnt(0);
    }
#endif
    __syncthreads();

    // w -> wk = MU + w^2, once per tile element
    for (int k = (int)threadIdx.x; k < HTY * HTX; k += NTHR) {
        float wv = sk_[k];
        sk_[k] = fmaf(wv, wv, MU_F);
    }
    __syncthreads();

    const float H2  = (1.0f / 1023.0f) * (1.0f / 1023.0f);
    const int   tx  = (int)threadIdx.x & (TX - 1);   // 0..63
    const int   ty0 = (int)threadIdx.x >> 6;         // 0..3
    const float* fb = f + (size_t)b * plane;

#pragma unroll
    for (int rr = 0; rr < TY / 4; ++rr) {
        const int r  = ty0 + rr * 4;        // local interior row 0..15
        const int lr = r + 1, lc = tx + 1;  // halo-tile coords

        float uN = su_[(lr - 1) * HTX + lc];
        float uS = su_[(lr + 1) * HTX + lc];
        float uW = su_[lr * HTX + lc - 1];
        float uE = su_[lr * HTX + lc + 1];
        float kN = sk_[(lr - 1) * HTX + lc];
        float kS = sk_[(lr + 1) * HTX + lc];
        float kW = sk_[lr * HTX + lc - 1];
        float kE = sk_[lr * HTX + lc + 1];
        float kc = sk_[lr * HTX + lc];

        float s_u = (uN + uS) + (uW + uE);                        // conv(u,k1)/0.5
        float s_v = fmaf(kN, uN, fmaf(kS, uS, fmaf(kW, uW, kE * uE))); // conv(wk*u,k2)/0.5
        float y3  = fmaf(0.5f, (kN + kS) + (kW + kE), 2.0f * kc); // conv(wk,k3)
        float fh  = fb[(size_t)(i0 + r + 1) * GDIM + (j0 + tx + 1)] * H2;
        float num = fmaf(0.5f, fmaf(s_u, kc, s_v), fh);
        float y   = num / y3;

        if (last) {
            out[(size_t)b * GINT * GINT + (size_t)(i0 + r) * GINT + (j0 + tx)] = y;
        } else {
            out[(size_t)b * plane + (size_t)(i0 + r + 1) * GDIM + (j0 + tx + 1)] = y;
        }
    }
}

// ---------------------------------------------------------------------------
extern "C" void kernel_launch(void* const* d_in, const int* in_sizes, int n_in,
                              void* d_out, int out_size, void* d_ws, size_t ws_size,
                              hipStream_t stream)
{
    (void)in_sizes; (void)n_in; (void)out_size; (void)ws_size;

    const float* pre = (const float*)d_in[0];  // (8,1,1022,1022)
    const float* f   = (const float*)d_in[1];  // (8,1,1024,1024)
    const float* w   = (const float*)d_in[2];  // (8,1,1024,1024)
    // d_in[3..5] = stencil kernels; values are compile-time constants here.

    float* u0 = (float*)d_ws;                          // 8 x 1024 x 1024 fp32
    float* u1 = u0 + (size_t)8 * GDIM * GDIM;          // ping-pong partner

    seed_kernel<<<(8 * GDIM * GDIM) / NTHR, NTHR, 0, stream>>>(pre, u0, u1);

    dim3 grid(GINT / TX + 1, GINT / TY + 1, 8);        // 16 x 64 x 8 tiles

    // 65 Jacobi sweeps total: jac_step(pre) + 64 fori_loop steps.
    for (int it = 0; it < 65; ++it) {
        const float* src = (it & 1) ? u1 : u0;
        float*       dst = (it & 1) ? u0 : u1;
        const int last = (it == 64);
        jac_kernel<<<grid, NTHR, 0, stream>>>(
            w, f, src, last ? (float*)d_out : dst, last);
    }
}